// LinkPredNet_51513837748919
// MI455X (gfx1250) — compile-verified
//
#include <hip/hip_runtime.h>
#include <hip/hip_bf16.h>

#define NNODES 16384
#define NEDGES 262144
#define INC    128
#define HIDC   128
#define OUTC   32

typedef __attribute__((ext_vector_type(2))) float v2f;
typedef __attribute__((ext_vector_type(8))) float v8f;

// D = A(16x4 f32) * B(4x16 f32) + C(16x16 f32)  -- native CDNA5 fp32 WMMA
__device__ __forceinline__ v8f wmma_f32x4(v2f a, v2f b, v8f c) {
  return __builtin_amdgcn_wmma_f32_16x16x4_f32(false, a, false, b, (short)0, c,
                                               false, false);
}

// fast sigmoid: v_exp_f32 + v_add + v_rcp_f32 (no IEEE divide expansion)
__device__ __forceinline__ float fast_sigmoid(float x) {
  return __builtin_amdgcn_rcpf(1.0f + __expf(-x));
}

// ---------------- degree / normalization ----------------
__global__ void k_deg_init(float* __restrict__ deg) {
  int i = blockIdx.x * blockDim.x + threadIdx.x;
  if (i < NNODES) deg[i] = 1.0f;  // self loop contributes 1
}

__global__ void k_deg_count(const long long* __restrict__ dst,
                            float* __restrict__ deg) {
  int e = blockIdx.x * blockDim.x + threadIdx.x;
  if (e < NEDGES) atomicAdd(&deg[(int)dst[e]], 1.0f);
}

__global__ void k_dinv(float* __restrict__ deg) {
  int i = blockIdx.x * blockDim.x + threadIdx.x;
  if (i < NNODES) deg[i] = rsqrtf(deg[i]);  // deg >= 1 always
}

// ---------------- dense GEMM: C[M,N] = (reluA? relu(A):A)[M,K] * B[K,N] ----
// one wave per 16x16 output tile; K-loop in steps of 4 with f32 WMMA.
// A-frag (16x4): lane l, reg j -> A[l%16][ (l/16)*2 + j ]
// B-frag (4x16): lane l, reg j -> B[ (l/16)*2 + j ][ l%16 ]
// C/D:           lane l, reg r -> C[ (l/16)*8 + r ][ l%16 ]
template <int KDIM, int NDIM, bool RELU_A>
__global__ void k_gemm_wmma(const float* __restrict__ A,
                            const float* __restrict__ B,
                            float* __restrict__ C) {
  constexpr int TN = NDIM / 16;
  const int gw   = blockIdx.x * (blockDim.x >> 5) + (threadIdx.x >> 5);
  const int lane = threadIdx.x & 31;
  const int tm = gw / TN;
  const int tn = gw % TN;
  const int mrow = tm * 16 + (lane & 15);
  const int kh   = (lane >> 4) << 1;  // 0 or 2
  const int ncol = tn * 16 + (lane & 15);

  v8f acc = {};
#pragma unroll 4
  for (int k0 = 0; k0 < KDIM; k0 += 4) {
    v2f a = *(const v2f*)&A[mrow * KDIM + k0 + kh];
    if (RELU_A) { a[0] = fmaxf(a[0], 0.0f); a[1] = fmaxf(a[1], 0.0f); }
    v2f b;
    b[0] = B[(k0 + kh)     * NDIM + ncol];
    b[1] = B[(k0 + kh + 1) * NDIM + ncol];
    acc = wmma_f32x4(a, b, acc);
  }

  const int row0 = tm * 16 + (lane >> 4) * 8;
#pragma unroll
  for (int r = 0; r < 8; ++r) C[(row0 + r) * NDIM + ncol] = acc[r];
}

// ---------------- GCN aggregation ----------------
// out[i][f] = xw[i][f]*dinv[i]^2 + bias[f]   (self-loop term + bias)
template <int F>
__global__ void k_agg_init(const float* __restrict__ xw,
                           const float* __restrict__ dinv,
                           const float* __restrict__ bias,
                           float* __restrict__ out) {
  int idx = blockIdx.x * blockDim.x + threadIdx.x;
  if (idx >= NNODES * F) return;
  int i = idx / F, f = idx % F;
  float di = dinv[i];
  out[idx] = xw[idx] * di * di + bias[f];
}

// out[dst] += xw[src] * dinv[src]*dinv[dst], one wave per edge
template <int F>
__global__ void k_agg_edges(const float* __restrict__ xw,
                            const float* __restrict__ dinv,
                            const long long* __restrict__ src,
                            const long long* __restrict__ dst,
                            float* __restrict__ out) {
  int e    = (blockIdx.x * blockDim.x + threadIdx.x) >> 5;
  int lane = threadIdx.x & 31;
  if (e >= NEDGES) return;
  int s = (int)src[e];
  int d = (int)dst[e];
  float norm = dinv[s] * dinv[d];
#pragma unroll
  for (int j = 0; j < F / 32; ++j) {
    int f = lane + j * 32;
    atomicAdd(&out[(size_t)d * F + f], xw[(size_t)s * F + f] * norm);
  }
}

// ---------------- adj = sigmoid(z @ z^T), z:[N,32] ----------------
// Register-blocked: each wave caches A-fragments for MT=4 M-tiles (64 rows)
// and sweeps TPC=16 N-tiles; every B-fragment load feeds 4 accumulators
// (4x B-reuse + 4 independent WMMA chains for latency hiding).
// B[k][n] = z[n][k]: same per-lane layout as an A-fragment on rows n.
__global__ void k_zzt_sigmoid(const float* __restrict__ z,
                              float* __restrict__ adj) {
  constexpr int K   = OUTC;                  // 32
  constexpr int MT  = 4;                     // M-tiles per wave
  constexpr int TPC = 16;                    // N-tiles per wave
  constexpr int NCH = (NNODES / 16) / TPC;   // 64 N-chunks

  const int gw   = blockIdx.x * (blockDim.x >> 5) + (threadIdx.x >> 5);
  const int lane = threadIdx.x & 31;
  const int mg    = gw / NCH;                // 0..255 (M-groups of 64 rows)
  const int chunk = gw % NCH;                // 0..63
  const int l15 = lane & 15;
  const int kh  = (lane >> 4) << 1;          // 0 or 2

  // cache A fragments for 4 M-tiles
  v2f afr[MT][K / 4];
#pragma unroll
  for (int m = 0; m < MT; ++m) {
    const int mrow = (mg * MT + m) * 16 + l15;
#pragma unroll
    for (int kk = 0; kk < K / 4; ++kk)
      afr[m][kk] = *(const v2f*)&z[mrow * K + kk * 4 + kh];
  }

  for (int t = 0; t < TPC; ++t) {
    const int nrow = (chunk * TPC + t) * 16 + l15;
    v2f bfr[K / 4];
#pragma unroll
    for (int kk = 0; kk < K / 4; ++kk)
      bfr[kk] = *(const v2f*)&z[nrow * K + kk * 4 + kh];

    v8f acc[MT];
#pragma unroll
    for (int m = 0; m < MT; ++m) acc[m] = (v8f){};
#pragma unroll
    for (int kk = 0; kk < K / 4; ++kk)
#pragma unroll
      for (int m = 0; m < MT; ++m)
        acc[m] = wmma_f32x4(afr[m][kk], bfr[kk], acc[m]);

#pragma unroll
    for (int m = 0; m < MT; ++m) {
      const int row0 = (mg * MT + m) * 16 + (lane >> 4) * 8;
#pragma unroll
      for (int r = 0; r < 8; ++r) {
        float v = fast_sigmoid(acc[m][r]);
        __builtin_nontemporal_store(v, &adj[(size_t)(row0 + r) * NNODES + nrow]);
      }
    }
  }
}

// ---------------- driver ----------------
extern "C" void kernel_launch(void* const* d_in, const int* in_sizes, int n_in,
                              void* d_out, int out_size, void* d_ws,
                              size_t ws_size, hipStream_t stream) {
  const float*     x  = (const float*)d_in[0];
  const long long* ei = (const long long*)d_in[1];  // [2, E] int64
  const float*     W1 = (const float*)d_in[2];
  const float*     b1 = (const float*)d_in[3];
  const float*     W2 = (const float*)d_in[4];
  const float*     b2 = (const float*)d_in[5];
  const long long* src = ei;
  const long long* dst = ei + NEDGES;

  float* adj = (float*)d_out;                            // [N, N]
  float* z   = (float*)d_out + (size_t)NNODES * NNODES;  // [N, 32]

  // workspace carve-up (all fp32, ~18 MB)
  float* dinv = (float*)d_ws;              // N
  float* xw1  = dinv + NNODES;             // N*128
  float* h    = xw1 + (size_t)NNODES*INC;  // N*128
  float* zw   = h   + (size_t)NNODES*HIDC; // N*32

  // 1) symmetric normalization
  k_deg_init<<<NNODES / 256, 256, 0, stream>>>(dinv);
  k_deg_count<<<NEDGES / 256, 256, 0, stream>>>(dst, dinv);
  k_dinv<<<NNODES / 256, 256, 0, stream>>>(dinv);

  // 2) xw1 = x @ W1   (M=16384, K=128, N=128)
  k_gemm_wmma<INC, HIDC, false>
      <<<(NNODES / 16) * (HIDC / 16) / 8, 256, 0, stream>>>(x, W1, xw1);

  // 3) h = aggregate(xw1) + b1
  k_agg_init<HIDC><<<NNODES * HIDC / 256, 256, 0, stream>>>(xw1, dinv, b1, h);
  k_agg_edges<HIDC><<<NEDGES * 32 / 256, 256, 0, stream>>>(xw1, dinv, src, dst, h);

  // 4) zw = relu(h) @ W2   (M=16384, K=128, N=32)
  k_gemm_wmma<HIDC, OUTC, true>
      <<<(NNODES / 16) * (OUTC / 16) / 8, 256, 0, stream>>>(h, W2, zw);

  // 5) z = aggregate(zw) + b2   (written straight into d_out tail)
  k_agg_init<OUTC><<<NNODES * OUTC / 256, 256, 0, stream>>>(zw, dinv, b2, z);
  k_agg_edges<OUTC><<<NEDGES * 32 / 256, 256, 0, stream>>>(zw, dinv, src, dst, z);

  // 6) adj = sigmoid(z @ z^T): (1024/4) M-groups x 64 chunks = 16384 waves
  k_zzt_sigmoid<<<(NNODES / 16 / 4) * ((NNODES / 16) / 16) / 8, 256, 0, stream>>>(z, adj);
}